// TwinsGCN_65060164599990
// MI455X (gfx1250) — compile-verified
//
#include <hip/hip_runtime.h>
#include <hip/hip_bf16.h>

typedef float v2f __attribute__((ext_vector_type(2)));
typedef float v8f __attribute__((ext_vector_type(8)));
typedef int   v4i_vs __attribute__((vector_size(16)));   // matches builtin param pointee

#define TPB 256

// ---------------- degree / normalization ----------------
__global__ void k_deg_init(float* __restrict__ deg, int n) {
  int i = blockIdx.x * blockDim.x + threadIdx.x;
  if (i < n) deg[i] = 1.0f;  // self-loop weight 1
}

__global__ void k_deg_acc(const int* __restrict__ col, const float* __restrict__ ew,
                          float* __restrict__ deg, int e) {
  int i = blockIdx.x * blockDim.x + threadIdx.x;
  if (i < e) atomicAdd(&deg[col[i]], ew[i]);
}

__global__ void k_deg_fin(float* __restrict__ deg, int n) {
  int i = blockIdx.x * blockDim.x + threadIdx.x;
  if (i < n) { float d = deg[i]; deg[i] = d > 0.f ? rsqrtf(d) : 0.f; }
}

// ---------------- fp32 WMMA GEMM: Out[N,C] = X[N,K] @ W[K,C] ----------------
// Block: 128 threads = 4 waves. Block tile: 64 rows x 64 cols.
// Wave w owns cols [colBase+16w, +16) and all 4 row sub-tiles (B frag reused 4x).
// A staged in LDS in K-chunks of 128 via async global->LDS copies (ASYNCcnt).
#define GEMM_KC 128
#define GEMM_KP (GEMM_KC + 4)   // padded row stride (floats); 528B = 16B-aligned

#if defined(__HIP_DEVICE_COMPILE__) && __has_builtin(__builtin_amdgcn_global_load_async_to_lds_b128)
#define ASYNC_LDS_COPY16(srcp, dstp)                                        \
  __builtin_amdgcn_global_load_async_to_lds_b128(                           \
      (__attribute__((address_space(1))) v4i_vs*)(                          \
          (__attribute__((address_space(1))) void*)((void*)(srcp))),        \
      (__attribute__((address_space(3))) v4i_vs*)(                          \
          (__attribute__((address_space(3))) void*)((void*)(dstp))),        \
      /*offset=*/0, /*cpol=*/0)
#else
#define ASYNC_LDS_COPY16(srcp, dstp) (*(float4*)(dstp) = *(const float4*)(srcp))
#endif

static __device__ __forceinline__ void gemm_async_drain() {
#if defined(__HIP_DEVICE_COMPILE__) && __has_builtin(__builtin_amdgcn_s_wait_asynccnt)
  __builtin_amdgcn_s_wait_asynccnt(0);
#endif
}

__global__ __launch_bounds__(128)
void k_gemm_wmma_f32(const float* __restrict__ X, const float* __restrict__ W,
                     float* __restrict__ Out, int nrows, int K, int C) {
  __shared__ float lds[64 * GEMM_KP];     // ~33 KB
  const int rowTile = blockIdx.x * 64;
  const int colBase = blockIdx.y * 64;
  const int tid  = threadIdx.x;
  const int wave = tid >> 5;
  const int lane = tid & 31;
  const int nloc  = lane & 15;            // A-frag row within sub-tile / B,D col
  const int khalf = (lane >> 4) << 1;     // lanes 0-15: K0/K1; lanes 16-31: K2/K3
  const int col   = colBase + wave * 16 + nloc;
  const bool fullTile = (rowTile + 64 <= nrows);

  v8f acc[4];
#pragma unroll
  for (int t = 0; t < 4; ++t) acc[t] = (v8f){0.f,0.f,0.f,0.f,0.f,0.f,0.f,0.f};

  const int vecsPerRow = GEMM_KC / 4;     // float4 chunks per staged row

  for (int k0 = 0; k0 < K; k0 += GEMM_KC) {
    __syncthreads();                      // previous chunk fully consumed

    // ---- stage A chunk [64 x GEMM_KC] into LDS (common case: no guards) ----
    if (fullTile) {
      for (int idx = tid; idx < 64 * vecsPerRow; idx += 128) {
        int r = idx / vecsPerRow;
        int v = idx - r * vecsPerRow;
        ASYNC_LDS_COPY16(X + (size_t)(rowTile + r) * K + k0 + v * 4,
                         &lds[r * GEMM_KP + v * 4]);
      }
    } else {
      for (int idx = tid; idx < 64 * vecsPerRow; idx += 128) {
        int r = idx / vecsPerRow;
        int v = idx - r * vecsPerRow;
        int gr = rowTile + r;
        int loff = r * GEMM_KP + v * 4;
        if (gr < nrows) {
          ASYNC_LDS_COPY16(X + (size_t)gr * K + k0 + v * 4, &lds[loff]);
        } else {
          *(float4*)&lds[loff] = make_float4(0.f, 0.f, 0.f, 0.f);
        }
      }
    }
    gemm_async_drain();
    __syncthreads();                      // staged tile visible to all waves

    // ---- 16 WMMAs per k-step group: 1 B frag reused across 4 row sub-tiles ----
    for (int k = 0; k < GEMM_KC; k += 4) {
      v2f b;                              // B: 4x16 fp32 fragment
      b.x = W[(size_t)(k0 + k + khalf) * C + col];
      b.y = W[(size_t)(k0 + k + khalf + 1) * C + col];
#pragma unroll
      for (int t = 0; t < 4; ++t) {
        v2f a = *(const v2f*)&lds[(t * 16 + nloc) * GEMM_KP + k + khalf];
        acc[t] = __builtin_amdgcn_wmma_f32_16x16x4_f32(
            /*neg_a=*/false, a, /*neg_b=*/false, b,
            /*c_mod=*/(short)0, acc[t], /*reuse_a=*/false, /*reuse_b=*/false);
      }
    }
  }

  // D layout: VGPR j -> M = j (lanes 0-15) / j+8 (lanes 16-31), N = lane%16
  if (fullTile) {
#pragma unroll
    for (int t = 0; t < 4; ++t) {
      const int mBase = rowTile + t * 16 + ((lane >> 4) << 3);
#pragma unroll
      for (int j = 0; j < 8; ++j)
        Out[(size_t)(mBase + j) * C + col] = acc[t][j];
    }
  } else {
#pragma unroll
    for (int t = 0; t < 4; ++t) {
      const int mBase = rowTile + t * 16 + ((lane >> 4) << 3);
#pragma unroll
      for (int j = 0; j < 8; ++j) {
        int m = mBase + j;
        if (m < nrows) Out[(size_t)m * C + col] = acc[t][j];
      }
    }
  }
}

// ---------------- normalized gather/scatter (GCN aggregation) ----------------
// Self loops first (also initializes accumulator): Acc[i,:] = dinv[i]^2 * XW[i,:]
__global__ void k_scatter_self(const float* __restrict__ dinv, const float* __restrict__ XW,
                               float* __restrict__ Acc, int n, int C) {
  long long idx = (long long)blockIdx.x * blockDim.x + threadIdx.x;
  long long tot = (long long)n * C;
  if (idx < tot) {
    int i = (int)(idx / C);
    float di = dinv[i];
    Acc[idx] = di * di * XW[idx];
  }
}

// Edges: Acc[c,:] += (dinv[r]*w*dinv[c]) * XW[r,:]   (L2-resident atomics)
__global__ void k_scatter_edges(const int* __restrict__ row, const int* __restrict__ col,
                                const float* __restrict__ ew, const float* __restrict__ dinv,
                                const float* __restrict__ XW, float* __restrict__ Acc,
                                int E, int C) {
  int e = blockIdx.x;
  int ch = blockIdx.y * blockDim.x + threadIdx.x;
  if (e >= E || ch >= C) return;
  int r = row[e], c = col[e];
  float coef = dinv[r] * ew[e] * dinv[c];
  atomicAdd(&Acc[(size_t)c * C + ch], coef * XW[(size_t)r * C + ch]);
}

// ---------------- bias + ReLU + LayerNorm (one block per node row) ----------------
__global__ __launch_bounds__(TPB)
void k_bias_relu_ln(const float* __restrict__ Z, const float* __restrict__ bias,
                    const float* __restrict__ g, const float* __restrict__ be,
                    float* __restrict__ Out, int C) {
  __shared__ float s1[TPB], s2[TPB];
  const int rowi = blockIdx.x;
  const int per = C / TPB;                 // 1 (C=256) or 2 (C=512)
  float local[2] = {0.f, 0.f};
  float sum = 0.f, sq = 0.f;
  for (int t = 0; t < per; ++t) {
    int ch = threadIdx.x + t * TPB;
    float v = Z[(size_t)rowi * C + ch] + bias[ch];
    v = fmaxf(v, 0.0f);
    local[t] = v; sum += v; sq += v * v;
  }
  s1[threadIdx.x] = sum; s2[threadIdx.x] = sq;
  __syncthreads();
  for (int off = TPB / 2; off > 0; off >>= 1) {
    if (threadIdx.x < off) { s1[threadIdx.x] += s1[threadIdx.x + off];
                             s2[threadIdx.x] += s2[threadIdx.x + off]; }
    __syncthreads();
  }
  float mean = s1[0] / C;
  float var  = s2[0] / C - mean * mean;
  float rstd = rsqrtf(var + 1e-5f);
  for (int t = 0; t < per; ++t) {
    int ch = threadIdx.x + t * TPB;
    Out[(size_t)rowi * C + ch] = (local[t] - mean) * rstd * g[ch] + be[ch];
  }
}

// ---------------- global mean pool + FC head ----------------
__global__ void k_pool_zero(float* __restrict__ pooled, float* __restrict__ cnt,
                            int total, int G) {
  int i = blockIdx.x * blockDim.x + threadIdx.x;
  if (i < total) pooled[i] = 0.f;
  if (i < G) cnt[i] = 0.f;
}

__global__ void k_pool_acc(const float* __restrict__ H, const int* __restrict__ batch,
                           float* __restrict__ pooled, float* __restrict__ cnt, int C) {
  int i = blockIdx.x;
  int ch = threadIdx.x;
  int b = batch[i];
  atomicAdd(&pooled[(size_t)b * C + ch], H[(size_t)i * C + ch]);
  if (ch == 0) atomicAdd(&cnt[b], 1.0f);
}

__global__ void k_fc_pool(const float* __restrict__ pooled, const float* __restrict__ cnt,
                          const float* __restrict__ Wfc, const float* __restrict__ bfc,
                          float* __restrict__ combined, int C, int OC, int stride, int colOff) {
  int gi = blockIdx.x;
  int oc = threadIdx.x;
  if (oc >= OC) return;
  float invc = 1.0f / fmaxf(cnt[gi], 1.0f);
  float s = bfc[oc];
  for (int k = 0; k < C; ++k)
    s += pooled[(size_t)gi * C + k] * invc * Wfc[(size_t)k * OC + oc];
  combined[(size_t)gi * stride + colOff + oc] = s;
}

__global__ void k_final_fc(const float* __restrict__ combined, const float* __restrict__ Wf,
                           const float* __restrict__ bf, float* __restrict__ out,
                           int G, int K, int CLS) {
  int idx = blockIdx.x * blockDim.x + threadIdx.x;
  if (idx >= G * CLS) return;
  int gi = idx / CLS, cc = idx - gi * CLS;
  float s = bf[cc];
  for (int k = 0; k < K; ++k)
    s += combined[(size_t)gi * K + k] * Wf[(size_t)k * CLS + cc];
  out[idx] = s;
}

// ---------------- host-side tower driver ----------------
static inline int ceil_div(int a, int b) { return (a + b - 1) / b; }

static void run_tower(const float* x, const int* ei, const float* ew, const int* batch,
                      const float* const* P,   // W1,b1,g1,be1,W2,b2,g2,be2,Wfc,bfc
                      float* dinv, float* bufA, float* bufB,
                      float* pooled, float* cnt, float* combined, int colOff,
                      int N, int E, int IN, int C1, int C2, int OC, int G,
                      hipStream_t s) {
  const int* row = ei;
  const int* col = ei + E;

  // symmetric normalization coefficients
  k_deg_init<<<ceil_div(N, TPB), TPB, 0, s>>>(dinv, N);
  k_deg_acc<<<ceil_div(E, TPB), TPB, 0, s>>>(col, ew, dinv, E);
  k_deg_fin<<<ceil_div(N, TPB), TPB, 0, s>>>(dinv, N);

  // conv1: bufA = x @ W1 ; bufB = scatter(bufA) ; bufA = LN(relu(bufB + b1))
  {
    dim3 g(ceil_div(N, 64), C1 / 64), b(128);
    k_gemm_wmma_f32<<<g, b, 0, s>>>(x, P[0], bufA, N, IN, C1);
  }
  k_scatter_self<<<ceil_div(N, TPB) * C1, TPB, 0, s>>>(dinv, bufA, bufB, N, C1);
  k_scatter_edges<<<dim3(E, C1 / TPB), TPB, 0, s>>>(row, col, ew, dinv, bufA, bufB, E, C1);
  k_bias_relu_ln<<<N, TPB, 0, s>>>(bufB, P[1], P[2], P[3], bufA, C1);

  // conv2: bufB = h1 @ W2 ; bufA = scatter(bufB) ; bufA = LN(relu(bufA + b2))
  {
    dim3 g(ceil_div(N, 64), C2 / 64), b(128);
    k_gemm_wmma_f32<<<g, b, 0, s>>>(bufA, P[4], bufB, N, C1, C2);
  }
  k_scatter_self<<<ceil_div(N, TPB) * C2, TPB, 0, s>>>(dinv, bufB, bufA, N, C2);
  k_scatter_edges<<<dim3(E, C2 / TPB), TPB, 0, s>>>(row, col, ew, dinv, bufB, bufA, E, C2);
  k_bias_relu_ln<<<N, TPB, 0, s>>>(bufA, P[5], P[6], P[7], bufA, C2);

  // global mean pool + per-tower FC into combined[:, colOff:colOff+OC]
  k_pool_zero<<<ceil_div(G * C2, TPB), TPB, 0, s>>>(pooled, cnt, G * C2, G);
  k_pool_acc<<<N, C2, 0, s>>>(bufA, batch, pooled, cnt, C2);
  k_fc_pool<<<G, OC, 0, s>>>(pooled, cnt, P[8], P[9], combined, C2, OC, 2 * OC, colOff);
}

extern "C" void kernel_launch(void* const* d_in, const int* in_sizes, int n_in,
                              void* d_out, int out_size, void* d_ws, size_t ws_size,
                              hipStream_t stream) {
  const float* x    = (const float*)d_in[0];
  const int*   ei_c = (const int*)  d_in[1];
  const float* ew_c = (const float*)d_in[2];
  const int*   ei_f = (const int*)  d_in[3];
  const float* ew_f = (const float*)d_in[4];
  const int*   bat  = (const int*)  d_in[5];
  const float* Pc[10], *Pf[10];
  for (int i = 0; i < 10; ++i) { Pc[i] = (const float*)d_in[6 + i];
                                 Pf[i] = (const float*)d_in[16 + i]; }
  const float* Wfin = (const float*)d_in[26];
  const float* bfin = (const float*)d_in[27];
  float* out = (float*)d_out;

  const int N   = in_sizes[5];              // 50000
  const int E   = in_sizes[2];              // 400000
  const int IN  = in_sizes[0] / N;          // 256
  const int C1  = in_sizes[7];              // 512 (b1)
  const int C2  = in_sizes[11];             // 256 (b2)
  const int OC  = in_sizes[15];             // 128 (bfc)
  const int CLS = in_sizes[27];             // 4
  const int G   = out_size / CLS;           // 16

  // workspace layout
  char* ws = (char*)d_ws;
  size_t o = 0;
  auto take = [&](size_t bytes) { char* p = ws + o; o += (bytes + 511) & ~(size_t)511; return p; };
  float* dinv     = (float*)take((size_t)N * sizeof(float));
  float* bufA     = (float*)take((size_t)N * C1 * sizeof(float));
  float* bufB     = (float*)take((size_t)N * C1 * sizeof(float));
  float* pooled   = (float*)take((size_t)G * C2 * sizeof(float));
  float* cnt      = (float*)take((size_t)G * sizeof(float));
  float* combined = (float*)take((size_t)G * 2 * OC * sizeof(float));
  (void)ws_size; (void)n_in;

  run_tower(x, ei_c, ew_c, bat, Pc, dinv, bufA, bufB, pooled, cnt, combined, 0,
            N, E, IN, C1, C2, OC, G, stream);
  run_tower(x, ei_f, ew_f, bat, Pf, dinv, bufA, bufB, pooled, cnt, combined, OC,
            N, E, IN, C1, C2, OC, G, stream);

  k_final_fc<<<1, 64, 0, stream>>>(combined, Wfin, bfin, out, G, 2 * OC, CLS);
}